// SRPCANet_25460566130775
// MI455X (gfx1250) — compile-verified
//
#include <hip/hip_runtime.h>
#include <math.h>

// ---------------- sizes ----------------
#define N1v 192
#define N2v 168
#define N3v 64
#define SN3v 128
#define NMv (N1v * N2v)       // 32256 rows when flattened
#define ELEM64 (NMv * N3v)    // 2064384
#define ELEM128 (NMv * SN3v)  // 4128768
#define NSv 168               // SVD matrix columns
#define GSZv (NSv * NSv)      // 28224
#define CHv 128               // channels (slim*n3)

typedef __attribute__((ext_vector_type(2))) float v2f;
typedef __attribute__((ext_vector_type(8))) float v8f;

// fp32 WMMA: D = A(16x4) * B(4x16) + C(16x16)
__device__ __forceinline__ v8f wmma_k4(v2f a, v2f b, v8f c) {
  return __builtin_amdgcn_wmma_f32_16x16x4_f32(false, a, false, b, (short)0, c,
                                               false, false);
}

// round-robin tournament pairing: round r, pair k, n even, n-1 rounds
__device__ __forceinline__ void pair_pq(int r, int k, int n, int& p, int& q) {
  const int m = n - 1;
  if (k == 0) {
    p = m;
    q = r % m;
  } else {
    p = (r + k) % m;
    int t = (r - k) % m;
    if (t < 0) t += m;
    q = t;
  }
}

// async global -> LDS (16B) and LDS -> global (16B), CDNA5 async-DMA path
__device__ __forceinline__ void async_g2l_b128(void* lds, const void* gaddr) {
  unsigned l = (unsigned)(uintptr_t)lds;
  asm volatile("global_load_async_to_lds_b128 %0, %1, off" ::"v"(l), "v"(gaddr)
               : "memory");
}
__device__ __forceinline__ void async_l2g_b128(void* gaddr, const void* lds) {
  unsigned l = (unsigned)(uintptr_t)lds;
  asm volatile("global_store_async_from_lds_b128 %0, %1, off" ::"v"(gaddr),
               "v"(l)
               : "memory");
}
__device__ __forceinline__ void wait_async0() {
  asm volatile("s_wait_asynccnt 0x0" ::: "memory");
}

// =========================================================================
// Elementwise kernels
// =========================================================================
__global__ void __launch_bounds__(256) k_copy(float* __restrict__ dst,
                                              const float* __restrict__ src,
                                              int n) {
  int i = blockIdx.x * 256 + threadIdx.x;
  if (i < n) dst[i] = src[i];
}

__global__ void __launch_bounds__(256) k_zero(float* __restrict__ dst, int n) {
  int i = blockIdx.x * 256 + threadIdx.x;
  if (i < n) dst[i] = 0.f;
}

// Xc[c][ki] = TL[ki][c] - Z1[ki][c]/g : LDS tile transpose, both sides
// coalesced. grid = (NMv/16)*(128/16) = 16128 blocks of 256.
__global__ void __launch_bounds__(256) k_prepXt(const float* __restrict__ TL,
                                                const float* __restrict__ Z1,
                                                const float* __restrict__ gptr,
                                                float* __restrict__ Xc) {
  __shared__ float tile[16][17];
  float invg = 1.f / gptr[0];
  int bc = blockIdx.x & 7;   // channel tile (8)
  int bk = blockIdx.x >> 3;  // ki tile (2016)
  int tx = threadIdx.x & 15, ty = threadIdx.x >> 4;
  size_t src = (size_t)(bk * 16 + ty) * 128 + bc * 16 + tx;
  tile[ty][tx] = TL[src] - Z1[src] * invg;
  __syncthreads();
  int ch2 = bc * 16 + ty;
  int ki2 = bk * 16 + tx;
  Xc[(size_t)ch2 * NMv + ki2] = tile[tx][ty];
}

// E = shrS(M - L - Z2/g, lam/g);  T2 = M - E - Z2/g   (ELEM64)
__global__ void __launch_bounds__(256) k_prepE(const float* __restrict__ M,
                                               const float* __restrict__ L,
                                               const float* __restrict__ Z2,
                                               const float* __restrict__ gptr,
                                               const float* __restrict__ lamptr,
                                               float* __restrict__ E,
                                               float* __restrict__ T2) {
  int i = blockIdx.x * 256 + threadIdx.x;
  if (i >= ELEM64) return;
  float invg = 1.f / gptr[0];
  float thr = lamptr[0] * invg;
  float z = Z2[i] * invg;
  float d = M[i] - L[i] - z;
  float t = fabsf(d) - thr;
  float e = (t > 0.f) ? copysignf(t, d) : 0.f;
  E[i] = e;
  T2[i] = M[i] - e - z;
}

// T[c][m][j] *= F[c][j]   (ELEM128 laid out [c][192][168])
__global__ void __launch_bounds__(256) k_scaleT(float* __restrict__ T,
                                                const float* __restrict__ F) {
  int i = blockIdx.x * 256 + threadIdx.x;
  if (i >= ELEM128) return;
  int c = i / (N1v * NSv);
  int j = i % NSv;
  T[i] *= F[c * NSv + j];
}

// Z1 += g*(N - TL)   (ELEM128)
__global__ void __launch_bounds__(256) k_updZ1(float* __restrict__ Z1,
                                               const float* __restrict__ Nb,
                                               const float* __restrict__ TL,
                                               const float* __restrict__ gptr) {
  int i = blockIdx.x * 256 + threadIdx.x;
  if (i >= ELEM128) return;
  float g = gptr[0];
  Z1[i] += g * (Nb[i] - TL[i]);
}

// Z2 += g*(L + E - M)   (ELEM64)
__global__ void __launch_bounds__(256) k_updZ2(float* __restrict__ Z2,
                                               const float* __restrict__ L,
                                               const float* __restrict__ E,
                                               const float* __restrict__ M,
                                               const float* __restrict__ gptr) {
  int i = blockIdx.x * 256 + threadIdx.x;
  if (i >= ELEM64) return;
  float g = gptr[0];
  Z2[i] += g * (L[i] + E[i] - M[i]);
}

// =========================================================================
// WMMA GEMM kernels (one 16x16 tile per wave32, 8 waves/block)
// =========================================================================

// TL[32256 x 128] = L[32256 x 64] @ W^T  (W is [128 x 64] row-major)
__global__ void __launch_bounds__(256) k_gemm_TL(const float* __restrict__ L,
                                                 const float* __restrict__ W,
                                                 float* __restrict__ TL) {
  const int lane = threadIdx.x & 31, wave = threadIdx.x >> 5;
  const int half = lane >> 4, l16 = lane & 15;
  int tile = blockIdx.x * 8 + wave;  // 2016*8 = 16128 tiles
  int tn = tile & 7, tm = tile >> 3;
  const int m0 = tm * 16, n0 = tn * 16;
  v8f acc = {};
  const float* Ap = L + (size_t)(m0 + l16) * 64;
  const float* Bp = W + (size_t)(n0 + l16) * 64;  // W^T[k][n] = W[n][k]
#pragma unroll
  for (int k0 = 0; k0 < 64; k0 += 4) {
    int ka = k0 + 2 * half;
    v2f a, b;
    a.x = Ap[ka];
    a.y = Ap[ka + 1];
    b.x = Bp[ka];
    b.y = Bp[ka + 1];
    acc = wmma_k4(a, b, acc);
  }
  float* Cp = TL + (size_t)(m0 + half * 8) * 128 + n0 + l16;
#pragma unroll
  for (int v = 0; v < 8; ++v) Cp[(size_t)v * 128] = acc[v];
}

// L[32256 x 64] = 0.5*(((N + Z1/g)[32256 x 128] @ W[128 x 64]) + T2)
__global__ void __launch_bounds__(256) k_temp1_L(const float* __restrict__ Nb,
                                                 const float* __restrict__ Z1,
                                                 const float* __restrict__ W,
                                                 const float* __restrict__ T2,
                                                 const float* __restrict__ gptr,
                                                 float* __restrict__ Lout) {
  const int lane = threadIdx.x & 31, wave = threadIdx.x >> 5;
  const int half = lane >> 4, l16 = lane & 15;
  float invg = 1.f / gptr[0];
  int tile = blockIdx.x * 8 + wave;  // 2016*4 = 8064 tiles
  int tn = tile & 3, tm = tile >> 2;
  const int m0 = tm * 16, n0 = tn * 16;
  v8f acc = {};
  const float* An = Nb + (size_t)(m0 + l16) * 128;
  const float* Az = Z1 + (size_t)(m0 + l16) * 128;
#pragma unroll 4
  for (int k0 = 0; k0 < 128; k0 += 4) {
    int ka = k0 + 2 * half;
    v2f a, b;
    a.x = An[ka] + Az[ka] * invg;
    a.y = An[ka + 1] + Az[ka + 1] * invg;
    b.x = W[(size_t)ka * 64 + n0 + l16];
    b.y = W[(size_t)(ka + 1) * 64 + n0 + l16];
    acc = wmma_k4(a, b, acc);
  }
#pragma unroll
  for (int v = 0; v < 8; ++v) {
    size_t idx = (size_t)(m0 + half * 8 + v) * 64 + n0 + l16;
    Lout[idx] = 0.5f * (acc[v] + T2[idx]);
  }
}

// G_c[168 x 168] = X_c^T @ X_c ; Xc layout [c][192][168], K = 192.
// A[i][k] = Xc[k*168+i] and B[k][j] = Xc[k*168+j]: lane-contiguous loads.
__global__ void __launch_bounds__(256) k_gram(const float* __restrict__ Xc,
                                              float* __restrict__ G) {
  const int lane = threadIdx.x & 31, wave = threadIdx.x >> 5;
  const int half = lane >> 4, l16 = lane & 15;
  int tile = blockIdx.x * 8 + wave;  // 128*121 = 15488 tiles
  int c = tile / 121;
  int t = tile % 121;
  int tm = t / 11, tn = t % 11;
  const int m0 = tm * 16, n0 = tn * 16;
  const int i = m0 + l16, j = n0 + l16;
  const float* Xp = Xc + (size_t)c * NMv;
  v8f acc = {};
#pragma unroll 4
  for (int k0 = 0; k0 < 192; k0 += 4) {
    int ka = k0 + 2 * half;
    v2f a, b;
    a.x = (i < NSv) ? Xp[(size_t)ka * NSv + i] : 0.f;
    a.y = (i < NSv) ? Xp[(size_t)(ka + 1) * NSv + i] : 0.f;
    b.x = (j < NSv) ? Xp[(size_t)ka * NSv + j] : 0.f;
    b.y = (j < NSv) ? Xp[(size_t)(ka + 1) * NSv + j] : 0.f;
    acc = wmma_k4(a, b, acc);
  }
  if (j < NSv) {
    float* Gc = G + (size_t)c * GSZv;
#pragma unroll
    for (int v = 0; v < 8; ++v) {
      int row = m0 + half * 8 + v;
      if (row < NSv) Gc[(size_t)row * NSv + j] = acc[v];
    }
  }
}

// T_c[192 x 168] = X_c[192 x 168] @ V_c[168 x 168] ; Xc channel-major
__global__ void __launch_bounds__(256) k_gemm_T(const float* __restrict__ Xc,
                                                const float* __restrict__ V,
                                                float* __restrict__ T) {
  const int lane = threadIdx.x & 31, wave = threadIdx.x >> 5;
  const int half = lane >> 4, l16 = lane & 15;
  int tile = blockIdx.x * 8 + wave;  // 128*132 = 16896 tiles
  int c = tile / 132;
  int t = tile % 132;
  int tm = t / 11, tn = t % 11;
  const int m0 = tm * 16, n0 = tn * 16;
  const int j = n0 + l16;
  const float* Vc = V + (size_t)c * GSZv;
  const float* Ap = Xc + (size_t)c * NMv + (size_t)(m0 + l16) * NSv;
  v8f acc = {};
#pragma unroll 6
  for (int k0 = 0; k0 < NSv; k0 += 4) {
    int ka = k0 + 2 * half;
    v2f a, b;
    a.x = Ap[ka];
    a.y = Ap[ka + 1];
    b.x = (j < NSv) ? Vc[(size_t)ka * NSv + j] : 0.f;
    b.y = (j < NSv) ? Vc[(size_t)(ka + 1) * NSv + j] : 0.f;
    acc = wmma_k4(a, b, acc);
  }
  if (j < NSv) {
    float* Tc = T + (size_t)c * N1v * NSv;
#pragma unroll
    for (int v = 0; v < 8; ++v) {
      int row = m0 + half * 8 + v;
      Tc[(size_t)row * NSv + j] = acc[v];
    }
  }
}

// N_c[192 x 168] = T_c @ V_c^T ; stored back as N[192][168][128]
__global__ void __launch_bounds__(256) k_gemm_N(const float* __restrict__ T,
                                                const float* __restrict__ V,
                                                float* __restrict__ Nb) {
  const int lane = threadIdx.x & 31, wave = threadIdx.x >> 5;
  const int half = lane >> 4, l16 = lane & 15;
  int tile = blockIdx.x * 8 + wave;  // 128*132 = 16896 tiles
  int c = tile / 132;
  int t = tile % 132;
  int tm = t / 11, tn = t % 11;
  const int m0 = tm * 16, n0 = tn * 16;
  const int j = n0 + l16;
  const float* Vc = V + (size_t)c * GSZv;
  const float* Tc = T + (size_t)c * N1v * NSv;
  const float* Ap = Tc + (size_t)(m0 + l16) * NSv;
  v8f acc = {};
#pragma unroll 6
  for (int k0 = 0; k0 < NSv; k0 += 4) {
    int ka = k0 + 2 * half;
    v2f a, b;
    a.x = Ap[ka];
    a.y = Ap[ka + 1];
    // B[k][n] = V^T[k][n] = V[n][k]
    b.x = (j < NSv) ? Vc[(size_t)j * NSv + ka] : 0.f;
    b.y = (j < NSv) ? Vc[(size_t)j * NSv + ka + 1] : 0.f;
    acc = wmma_k4(a, b, acc);
  }
  if (j < NSv) {
#pragma unroll
    for (int v = 0; v < 8; ++v) {
      int row = m0 + half * 8 + v;
      Nb[((size_t)row * NSv + j) * 128 + c] = acc[v];
    }
  }
}

// =========================================================================
// Parallel cyclic Jacobi eigensolver: 168x168 symmetric, A + V in LDS
// (2 * 168*168*4 B = 226 KB of CDNA5's 320 KB per-workgroup LDS).
// G is DMA'd in with global_load_async_to_lds_b128 (overlapped with V=I),
// V is DMA'd out with global_store_async_from_lds_b128.
// =========================================================================
__global__ void __launch_bounds__(256)
    k_eigen(const float* __restrict__ G, const float* __restrict__ gptr,
            float* __restrict__ Vout, float* __restrict__ Fout) {
  const int c = blockIdx.x;
  const int tid = threadIdx.x;
  extern __shared__ float smem[];
  float* A = smem;         // 168*168
  float* V = smem + GSZv;  // 168*168
  __shared__ float rc[84], rs[84];

  const float* Gc = G + (size_t)c * GSZv;
  // async DMA G -> LDS (7056 x b128), overlapped with V init
  for (int i = tid; i < GSZv / 4; i += 256)
    async_g2l_b128(A + i * 4, Gc + i * 4);
  for (int i = tid; i < GSZv; i += 256) V[i] = 0.f;
  wait_async0();
  __syncthreads();
  for (int i = tid; i < NSv; i += 256) V[i * NSv + i] = 1.f;
  __syncthreads();

  const int n = NSv, rounds = NSv - 1, npairs = NSv / 2;
  for (int sweep = 0; sweep < 8; ++sweep) {
    for (int r = 0; r < rounds; ++r) {
      if (tid < npairs) {
        int p, q;
        pair_pq(r, tid, n, p, q);
        float app = A[p * n + p], aqq = A[q * n + q], apq = A[p * n + q];
        float cth = 1.f, sth = 0.f;
        if (fabsf(apq) > 1e-20f) {
          float tau = (aqq - app) / (2.f * apq);
          float tt =
              copysignf(1.f, tau) / (fabsf(tau) + sqrtf(1.f + tau * tau));
          cth = 1.f / sqrtf(1.f + tt * tt);
          sth = tt * cth;
        }
        rc[tid] = cth;
        rs[tid] = sth;
      }
      __syncthreads();
      // column update: A <- A*J, V <- V*J (disjoint column pairs)
      for (int w = tid; w < npairs * n; w += 256) {
        int k = w / n, row = w % n;
        int p, q;
        pair_pq(r, k, n, p, q);
        float cth = rc[k], sth = rs[k];
        float ap = A[row * n + p], aq = A[row * n + q];
        A[row * n + p] = cth * ap - sth * aq;
        A[row * n + q] = sth * ap + cth * aq;
        float vp = V[row * n + p], vq = V[row * n + q];
        V[row * n + p] = cth * vp - sth * vq;
        V[row * n + q] = sth * vp + cth * vq;
      }
      __syncthreads();
      // row update: A <- J^T * A
      for (int w = tid; w < npairs * n; w += 256) {
        int k = w / n, col = w % n;
        int p, q;
        pair_pq(r, k, n, p, q);
        float cth = rc[k], sth = rs[k];
        float ap = A[p * n + col], aq = A[q * n + col];
        A[p * n + col] = cth * ap - sth * aq;
        A[q * n + col] = sth * ap + cth * aq;
      }
      __syncthreads();
    }
  }

  // async DMA V -> global
  float* Vc = Vout + (size_t)c * GSZv;
  for (int i = tid; i < GSZv / 4; i += 256)
    async_l2g_b128(Vc + i * 4, V + i * 4);
  float g = gptr[0];
  float thr = 1.f / (g * sqrtf(128.f));
  for (int j2 = tid; j2 < NSv; j2 += 256) {
    float lam = A[j2 * n + j2];
    float s = sqrtf(fmaxf(lam, 0.f));
    Fout[c * NSv + j2] = (s > thr) ? (s - thr) / s : 0.f;
  }
  wait_async0();
}

// =========================================================================
// loss_orth = cond(W) - 1 via Jacobi eigen of W^T W (64x64) in LDS
// =========================================================================
__global__ void __launch_bounds__(256) k_loss(const float* __restrict__ W,
                                              float* __restrict__ out) {
  __shared__ float A[64 * 64];
  __shared__ float rc[32], rs[32];
  const int tid = threadIdx.x;
  for (int t = tid; t < 4096; t += 256) {
    int i = t >> 6, j = t & 63;
    float s = 0.f;
    for (int k = 0; k < 128; ++k) s += W[k * 64 + i] * W[k * 64 + j];
    A[t] = s;
  }
  __syncthreads();
  const int n = 64, rounds = 63, npairs = 32;
  for (int sweep = 0; sweep < 12; ++sweep) {
    for (int r = 0; r < rounds; ++r) {
      if (tid < npairs) {
        int p, q;
        pair_pq(r, tid, n, p, q);
        float app = A[p * n + p], aqq = A[q * n + q], apq = A[p * n + q];
        float cth = 1.f, sth = 0.f;
        if (fabsf(apq) > 1e-20f) {
          float tau = (aqq - app) / (2.f * apq);
          float tt =
              copysignf(1.f, tau) / (fabsf(tau) + sqrtf(1.f + tau * tau));
          cth = 1.f / sqrtf(1.f + tt * tt);
          sth = tt * cth;
        }
        rc[tid] = cth;
        rs[tid] = sth;
      }
      __syncthreads();
      for (int w = tid; w < npairs * n; w += 256) {
        int k = w / n, row = w % n;
        int p, q;
        pair_pq(r, k, n, p, q);
        float cth = rc[k], sth = rs[k];
        float ap = A[row * n + p], aq = A[row * n + q];
        A[row * n + p] = cth * ap - sth * aq;
        A[row * n + q] = sth * ap + cth * aq;
      }
      __syncthreads();
      for (int w = tid; w < npairs * n; w += 256) {
        int k = w / n, col = w % n;
        int p, q;
        pair_pq(r, k, n, p, q);
        float cth = rc[k], sth = rs[k];
        float ap = A[p * n + col], aq = A[q * n + col];
        A[p * n + col] = cth * ap - sth * aq;
        A[q * n + col] = sth * ap + cth * aq;
      }
      __syncthreads();
    }
  }
  if (tid == 0) {
    float mx = 0.f, mn = 3.4e38f;
    for (int i = 0; i < 64; ++i) {
      float l = fmaxf(A[i * 64 + i], 0.f);
      mx = fmaxf(mx, l);
      mn = fminf(mn, l);
    }
    out[0] = sqrtf(mx / fmaxf(mn, 1e-30f)) - 1.f;
  }
}

// =========================================================================
// host launcher
// =========================================================================
extern "C" void kernel_launch(void* const* d_in, const int* in_sizes, int n_in,
                              void* d_out, int out_size, void* d_ws,
                              size_t ws_size, hipStream_t stream) {
  const float* Mv = (const float*)d_in[0];
  const float* lam = (const float*)d_in[1];
  const float* Linit = (const float*)d_in[2];
  const float* W = (const float*)d_in[3];
  const float* gamma = (const float*)d_in[4];
  float* out = (float*)d_out;

  float* ws = (float*)d_ws;
  size_t off = 0;
  float* TL = ws + off; off += (size_t)ELEM128;
  float* Z1 = ws + off; off += (size_t)ELEM128;
  float* Z2 = ws + off; off += (size_t)ELEM64;
  float* Lb = ws + off; off += (size_t)ELEM64;
  float* Nb = ws + off; off += (size_t)ELEM128;
  float* Eb = ws + off; off += (size_t)ELEM64;
  float* T2 = ws + off; off += (size_t)ELEM64;
  float* Xc = ws + off; off += (size_t)ELEM128;  // channel-major X
  float* Gb = ws + off; off += (size_t)CHv * GSZv;
  float* Vb = ws + off; off += (size_t)CHv * GSZv;
  float* Fb = ws + off; off += (size_t)CHv * NSv;
  float* Tb = ws + off; off += (size_t)ELEM128;

  const int g128 = ELEM128 / 256;  // 16128
  const int g64 = ELEM64 / 256;    // 8064

  // init: L = L_init, Z1 = Z2 = 0, TL = L @ W^T
  k_copy<<<g64, 256, 0, stream>>>(Lb, Linit, ELEM64);
  k_zero<<<g128, 256, 0, stream>>>(Z1, ELEM128);
  k_zero<<<g64, 256, 0, stream>>>(Z2, ELEM64);
  k_gemm_TL<<<2016, 256, 0, stream>>>(Lb, W, TL);

  const size_t eig_lds = 2ull * GSZv * sizeof(float);  // 226 KB
  for (int s = 0; s < 9; ++s) {
    const float* g = gamma + s;
    k_prepXt<<<16128, 256, 0, stream>>>(TL, Z1, g, Xc);
    k_prepE<<<g64, 256, 0, stream>>>(Mv, Lb, Z2, g, lam, Eb, T2);
    k_gram<<<1936, 256, 0, stream>>>(Xc, Gb);
    k_eigen<<<128, 256, eig_lds, stream>>>(Gb, g, Vb, Fb);
    k_gemm_T<<<2112, 256, 0, stream>>>(Xc, Vb, Tb);
    k_scaleT<<<g128, 256, 0, stream>>>(Tb, Fb);
    k_gemm_N<<<2112, 256, 0, stream>>>(Tb, Vb, Nb);
    k_temp1_L<<<1008, 256, 0, stream>>>(Nb, Z1, W, T2, g, Lb);
    k_gemm_TL<<<2016, 256, 0, stream>>>(Lb, W, TL);
    k_updZ1<<<g128, 256, 0, stream>>>(Z1, Nb, TL, g);
    k_updZ2<<<g64, 256, 0, stream>>>(Z2, Lb, Eb, Mv, g);
  }

  k_copy<<<g64, 256, 0, stream>>>(out, Lb, ELEM64);
  k_loss<<<1, 256, 0, stream>>>(W, out + ELEM64);
}